// GNN_5076651344578
// MI455X (gfx1250) — compile-verified
//
#include <hip/hip_runtime.h>

// ---------------------------------------------------------------------------
// GNN (2x GCNConv + dueling-head MLP) for MI455X / gfx1250.
// Dense 32x32 matmuls use V_WMMA_F32_16X16X4_F32 (full fp32, matches ref).
// Edge scatter: one wave32 per edge, lane = feature dim (coalesced L2 atomics).
// ---------------------------------------------------------------------------

typedef __attribute__((ext_vector_type(2))) float v2f;
typedef __attribute__((ext_vector_type(8))) float v8f;

#define DIM 32
#define TLDA 33   // padded LDS stride for 16x32 tiles (bank-conflict free A reads)

// ---------------------------------------------------------------------------
// One wave computes C[16][32] = A[16][32] * B[32][32] using 16 fp32 WMMAs.
//   sA : 16 x 32 row-major, stride TLDA
//   sB : 32 x 32, sB[k*32+n] = W[n][k]   (i.e. W^T, so out = in @ W^T)
// A layout (16x4 f32): lanes 0-15 hold {K=k0,K=k0+1} of row m, lanes 16-31
// hold {K=k0+2,K=k0+3}. B mirrors with N=lane%16. C: v8f, row = i+8*(lane/16),
// col = lane%16 (+16 for acc1).
// ---------------------------------------------------------------------------
__device__ __forceinline__ void wave_mm_16x32x32(const float* __restrict__ sA,
                                                 const float* __restrict__ sB,
                                                 int lane, v8f& acc0, v8f& acc1)
{
    const int hb = lane >> 4;    // half of wave
    const int m  = lane & 15;
#pragma unroll
    for (int kk = 0; kk < 8; ++kk) {
        const int k0 = kk * 4 + 2 * hb;
        v2f a, b0, b1;
        a.x  = sA[m * TLDA + k0];
        a.y  = sA[m * TLDA + k0 + 1];
        b0.x = sB[k0 * 32 + m];
        b0.y = sB[(k0 + 1) * 32 + m];
        b1.x = sB[k0 * 32 + 16 + m];
        b1.y = sB[(k0 + 1) * 32 + 16 + m];
        acc0 = __builtin_amdgcn_wmma_f32_16x16x4_f32(false, a, false, b0,
                                                     (short)0, acc0, false, false);
        acc1 = __builtin_amdgcn_wmma_f32_16x16x4_f32(false, a, false, b1,
                                                     (short)0, acc1, false, false);
    }
}

// ------------------------- small elementwise kernels -----------------------
__global__ void k_init_deg(float* deg, int n) {
    int i = blockIdx.x * blockDim.x + threadIdx.x;
    if (i < n) deg[i] = 1.0f;                       // self-loop
}

__global__ void k_deg_accum(const int* __restrict__ dst, float* __restrict__ deg, int e) {
    int i = blockIdx.x * blockDim.x + threadIdx.x;
    if (i < e) atomicAdd(&deg[dst[i]], 1.0f);
}

__global__ void k_rsqrt(float* deg, int n) {
    int i = blockIdx.x * blockDim.x + threadIdx.x;
    if (i < n) deg[i] = rsqrtf(deg[i]);
}

// acc[n,d] = xw[n,d] * dis[n]^2   (self-loop contribution)
__global__ void k_scatter_init(const float* __restrict__ xw, const float* __restrict__ dis,
                               float* __restrict__ acc, int n32) {
    int i = blockIdx.x * blockDim.x + threadIdx.x;
    if (i < n32) { float s = dis[i >> 5]; acc[i] = xw[i] * s * s; }
}

// one wave per edge; lane = feature dim
__global__ void k_scatter_edges(const float* __restrict__ xw, const float* __restrict__ dis,
                                const int* __restrict__ src, const int* __restrict__ dst,
                                float* __restrict__ acc, int e) {
    unsigned long long idx = (unsigned long long)blockIdx.x * blockDim.x + threadIdx.x;
    int ei = (int)(idx >> 5);
    int d  = (int)(idx & 31);
    if (ei < e) {
        int s = src[ei], t = dst[ei];
        float norm = dis[s] * dis[t];
        atomicAdd(&acc[(unsigned long long)t * DIM + d], xw[(unsigned long long)s * DIM + d] * norm);
    }
}

__global__ void k_bias_relu(const float* __restrict__ acc, const float* __restrict__ b,
                            float* __restrict__ h, int n32) {
    int i = blockIdx.x * blockDim.x + threadIdx.x;
    if (i < n32) h[i] = fmaxf(acc[i] + b[i & 31], 0.0f);
}

// ------------------------- WMMA linear: out = in @ W^T ---------------------
#define LIN_WAVES 4
__global__ void k_linear_wmma(const float* __restrict__ in, const float* __restrict__ W,
                              float* __restrict__ out, int nRows)
{
    __shared__ float sW[32 * 32];
    __shared__ float sA[LIN_WAVES][16 * TLDA];
    const int tid  = threadIdx.x;        // 128 threads = 4 waves
    const int wave = tid >> 5;
    const int lane = tid & 31;

    for (int i = tid; i < 1024; i += 128) {       // sW[k*32+n] = W[n][k]
        int k = i >> 5, n = i & 31;
        sW[i] = W[n * 32 + k];
    }
    const int base = (blockIdx.x * LIN_WAVES + wave) * 16;
    const bool act = (base < nRows);              // wave-uniform (rows % 16 == 0)
    if (act) {
        for (int i = lane; i < 16 * DIM; i += 32)
            sA[wave][(i >> 5) * TLDA + (i & 31)] = in[(unsigned long long)base * DIM + i];
    }
    __syncthreads();
    if (act) {
        v8f a0 = {}, a1 = {};
        wave_mm_16x32x32(sA[wave], sW, lane, a0, a1);
        const int hb = lane >> 4, m = lane & 15;
#pragma unroll
        for (int i = 0; i < 8; ++i) {
            unsigned long long row = base + i + 8 * hb;
            out[row * DIM + m]      = a0[i];
            out[row * DIM + 16 + m] = a1[i];
        }
    }
}

// ------------------- fused triple-gather + dueling MLP ---------------------
#define MLP_WAVES 4
__global__ void k_mlp(const float* __restrict__ h,
                      const int* __restrict__ head, const int* __restrict__ tail,
                      const int* __restrict__ rel,  const float* __restrict__ rel_emb,
                      const float* __restrict__ aw1, const float* __restrict__ ab1,
                      const float* __restrict__ aw2, const float* __restrict__ ab2,
                      const float* __restrict__ aw3, const float* __restrict__ ab3,
                      const float* __restrict__ vw1, const float* __restrict__ vb1,
                      const float* __restrict__ vw2, const float* __restrict__ vb2,
                      const float* __restrict__ vw3, const float* __restrict__ vb3,
                      float* __restrict__ q, int T)
{
    __shared__ float sW[4][1024];                 // aw1, aw2, vw1, vw2 (as W^T)
    __shared__ float sT[MLP_WAVES][16 * TLDA];    // triple tile
    __shared__ float sH1[MLP_WAVES][16 * TLDA];   // hidden scratch
    __shared__ float sH2[MLP_WAVES][16 * TLDA];   // adv hidden (preserved)

    const int tid  = threadIdx.x;                 // 128 threads
    const int wave = tid >> 5;
    const int lane = tid & 31;
    const int hb   = lane >> 4, m = lane & 15;

    const float* Ws[4] = {aw1, aw2, vw1, vw2};
    for (int i = tid; i < 4096; i += 128) {
        int w = i >> 10, r = i & 1023;
        int k = r >> 5, n = r & 31;
        sW[w][r] = Ws[w][n * 32 + k];
    }
    const int tile = blockIdx.x * MLP_WAVES + wave;
    const int base = tile * 16;
    const bool act = (base < T);                  // wave-uniform (T % 16 == 0)
    if (act) {
        for (int it = 0; it < 16; ++it) {         // all 32 lanes: one row/iter
            int t = base + it;
            int hn = head[t], tn = tail[t], rn = rel[t];
            sT[wave][it * TLDA + lane] =
                h[(unsigned long long)hn * DIM + lane] +
                h[(unsigned long long)tn * DIM + lane] +
                rel_emb[(unsigned long long)rn * DIM + lane];
        }
    }
    __syncthreads();

    auto stage = [&](const float* src, const float* sB, const float* bias, float* dst) {
        v8f a0 = {}, a1 = {};
        if (act) wave_mm_16x32x32(src, sB, lane, a0, a1);
        __syncthreads();                          // src may alias dst
        if (act) {
            float b0 = bias[m], b1 = bias[16 + m];
#pragma unroll
            for (int i = 0; i < 8; ++i) {
                dst[(i + 8 * hb) * TLDA + m]      = fmaxf(a0[i] + b0, 0.0f);
                dst[(i + 8 * hb) * TLDA + 16 + m] = fmaxf(a1[i] + b1, 0.0f);
            }
        }
        __syncthreads();
    };

    stage(sT[wave],  sW[0], ab1, sH1[wave]);   // adv layer 1
    stage(sH1[wave], sW[1], ab2, sH2[wave]);   // adv layer 2
    stage(sT[wave],  sW[2], vb1, sH1[wave]);   // val layer 1
    stage(sH1[wave], sW[3], vb2, sH1[wave]);   // val layer 2 (in-place, barriered)

    if (act && lane < 16) {                    // tiny 32->3 / 32->1 heads
        int t = base + lane;
        float a0 = ab3[0], a1 = ab3[1], a2 = ab3[2], v = vb3[0];
        for (int k = 0; k < 32; ++k) {
            float hA = sH2[wave][lane * TLDA + k];
            float hV = sH1[wave][lane * TLDA + k];
            a0 += hA * aw3[0 * 32 + k];
            a1 += hA * aw3[1 * 32 + k];
            a2 += hA * aw3[2 * 32 + k];
            v  += hV * vw3[k];
        }
        float mean = (a0 + a1 + a2) * (1.0f / 3.0f);
        q[t * 3 + 0] = v + a0 - mean;
        q[t * 3 + 1] = v + a1 - mean;
        q[t * 3 + 2] = v + a2 - mean;
    }
}

// ---------------------------------------------------------------------------
extern "C" void kernel_launch(void* const* d_in, const int* in_sizes, int n_in,
                              void* d_out, int out_size, void* d_ws, size_t ws_size,
                              hipStream_t stream)
{
    const float* x       = (const float*)d_in[0];
    const int*   ei      = (const int*)  d_in[1];
    const int*   head    = (const int*)  d_in[2];
    const int*   tail    = (const int*)  d_in[3];
    const int*   relidx  = (const int*)  d_in[4];
    const float* rel_emb = (const float*)d_in[5];
    const float* gw1 = (const float*)d_in[6];  const float* gb1 = (const float*)d_in[7];
    const float* gw2 = (const float*)d_in[8];  const float* gb2 = (const float*)d_in[9];
    const float* aw1 = (const float*)d_in[10]; const float* ab1 = (const float*)d_in[11];
    const float* aw2 = (const float*)d_in[12]; const float* ab2 = (const float*)d_in[13];
    const float* aw3 = (const float*)d_in[14]; const float* ab3 = (const float*)d_in[15];
    const float* vw1 = (const float*)d_in[16]; const float* vb1 = (const float*)d_in[17];
    const float* vw2 = (const float*)d_in[18]; const float* vb2 = (const float*)d_in[19];
    const float* vw3 = (const float*)d_in[20]; const float* vb3 = (const float*)d_in[21];

    const int N = in_sizes[0] / DIM;
    const int E = in_sizes[1] / 2;
    const int T = in_sizes[2];
    const int* src = ei;
    const int* dst = ei + E;

    float* wsf  = (float*)d_ws;
    float* dis  = wsf;                                   // [N]
    float* bufA = dis  + N;                              // [N*32]  xw
    float* bufB = bufA + (size_t)N * DIM;                // [N*32]  scatter accum
    float* bufC = bufB + (size_t)N * DIM;                // [N*32]  h

    const int n32 = N * DIM;
    const unsigned long long e32 = (unsigned long long)E * DIM;
    dim3 blk(256);
    dim3 gN((N + 255) / 256), gE((E + 255) / 256), gN32((n32 + 255) / 256);
    dim3 gE32((unsigned)((e32 + 255) / 256));
    dim3 blkW(128);
    dim3 gLin((N / 16 + LIN_WAVES - 1) / LIN_WAVES);
    dim3 gMlp(((T + 15) / 16 + MLP_WAVES - 1) / MLP_WAVES);

    // degrees -> dis = rsqrt(deg)
    k_init_deg<<<gN, blk, 0, stream>>>(dis, N);
    k_deg_accum<<<gE, blk, 0, stream>>>(dst, dis, E);
    k_rsqrt<<<gN, blk, 0, stream>>>(dis, N);

    // GCN layer 1
    k_linear_wmma<<<gLin, blkW, 0, stream>>>(x, gw1, bufA, N);
    k_scatter_init<<<gN32, blk, 0, stream>>>(bufA, dis, bufB, n32);
    k_scatter_edges<<<gE32, blk, 0, stream>>>(bufA, dis, src, dst, bufB, E);
    k_bias_relu<<<gN32, blk, 0, stream>>>(bufB, gb1, bufC, n32);

    // GCN layer 2
    k_linear_wmma<<<gLin, blkW, 0, stream>>>(bufC, gw2, bufA, N);
    k_scatter_init<<<gN32, blk, 0, stream>>>(bufA, dis, bufB, n32);
    k_scatter_edges<<<gE32, blk, 0, stream>>>(bufA, dis, src, dst, bufB, E);
    k_bias_relu<<<gN32, blk, 0, stream>>>(bufB, gb2, bufC, n32);

    // fused triple gather + dueling MLP -> q [T,3]
    k_mlp<<<gMlp, blkW, 0, stream>>>(bufC, head, tail, relidx, rel_emb,
                                     aw1, ab1, aw2, ab2, aw3, ab3,
                                     vw1, vb1, vw2, vb2, vw3, vb3,
                                     (float*)d_out, T);
    (void)n_in; (void)out_size; (void)ws_size;
}